// SelectiveElmanCell_438086664568
// MI455X (gfx1250) — compile-verified
//
#include <hip/hip_runtime.h>
#include <stdint.h>
#include <stddef.h>

// Problem constants (fixed by the reference setup_inputs()).
#define T_STEPS 2048
#define BATCH   16
#define DIM     1024
#define KMEM    4
#define NBLK    64          // persistent scan blocks; 64 * 16 cols = DIM
#define TARGET_RADIUS 0.99f
#define EPSV    1e-8f

// LDS layout for the scan kernel: W block padded to 1032 dwords/row
#define WROW    1032
#define SCAN_SMEM_FLOATS (16*WROW + 8*256 + KMEM*256)
#define SCAN_SMEM_BYTES  (SCAN_SMEM_FLOATS*4)

typedef __attribute__((ext_vector_type(2)))  float        v2f;
typedef __attribute__((ext_vector_type(8)))  float        v8f;
typedef __attribute__((ext_vector_type(4)))  unsigned int v4u;
typedef __attribute__((ext_vector_type(8)))  int          v8i;
typedef __attribute__((ext_vector_type(4)))  int          v4i;

__device__ __forceinline__ float sigmoidf_(float x) { return 1.0f / (1.0f + expf(-x)); }
__device__ __forceinline__ float siluf_(float x)    { return x / (1.0f + expf(-x)); }

// ---------------------------------------------------------------------------
// Block-wide sum reduction (blockDim.x == 1024)
// ---------------------------------------------------------------------------
__device__ __forceinline__ float block_reduce_sum(float x, float* red) {
    const int tid = threadIdx.x;
    red[tid] = x;
    __syncthreads();
    for (int s = 512; s > 0; s >>= 1) {
        if (tid < s) red[tid] += red[tid + s];
        __syncthreads();
    }
    float r = red[0];
    __syncthreads();
    return r;
}

// ---------------------------------------------------------------------------
// Kernel 1: spectral-norm scale for W_h (single block, 1024 threads).
// 3 power iterations; sigma = |u . (W v)| with final u,v. Deterministic
// hash-based normal init for u0 (JAX threefry not reproducible on device;
// power iteration is insensitive to the seed to within its own tolerance).
// ---------------------------------------------------------------------------
__global__ __launch_bounds__(1024) void k_spectral(const float* __restrict__ Wh,
                                                   float* __restrict__ scale_out) {
    __shared__ float u[DIM];
    __shared__ float v[DIM];
    __shared__ float red[1024];
    const int tid = threadIdx.x;

    unsigned h1 = (unsigned)tid * 2654435761u + 0x9E3779B9u;
    h1 ^= h1 >> 16; h1 *= 0x85EBCA6Bu; h1 ^= h1 >> 13; h1 *= 0xC2B2AE35u; h1 ^= h1 >> 16;
    unsigned h2 = h1 * 0x27D4EB2Du + 0x165667B1u;
    h2 ^= h2 >> 15; h2 *= 0x9E3779B1u; h2 ^= h2 >> 16;
    float u1 = ((float)(h1 & 0xFFFFFFu) + 1.0f) * (1.0f / 16777216.0f);
    float u2 = (float)(h2 & 0xFFFFFFu) * (1.0f / 16777216.0f);
    float g  = sqrtf(-2.0f * logf(u1)) * cosf(6.28318530718f * u2);

    float nrm = sqrtf(block_reduce_sum(g * g, red));
    u[tid] = g / nrm;
    __syncthreads();

    float t_last = 0.0f;
    for (int it = 0; it < 3; ++it) {
        // v = normalize(W^T u)   (thread tid owns column tid; coalesced rows)
        float acc = 0.0f;
        for (int i = 0; i < DIM; ++i) acc += Wh[(size_t)i * DIM + tid] * u[i];
        float vn = sqrtf(block_reduce_sum(acc * acc, red));
        v[tid] = acc / (vn + EPSV);
        __syncthreads();
        // u = normalize(W v)     (thread tid owns row tid)
        acc = 0.0f;
        for (int j = 0; j < DIM; ++j) acc += Wh[(size_t)tid * DIM + j] * v[j];
        float un = sqrtf(block_reduce_sum(acc * acc, red));
        t_last = acc;                 // unnormalized W v of the last iteration
        u[tid] = acc / (un + EPSV);
        __syncthreads();
    }
    float sigma = fabsf(block_reduce_sum(u[tid] * t_last, red));
    if (tid == 0) scale_out[0] = TARGET_RADIUS / (sigma + EPSV);
}

// ---------------------------------------------------------------------------
// Kernel 2: W_hn = W_h * scale
// ---------------------------------------------------------------------------
__global__ __launch_bounds__(256) void k_scale(const float* __restrict__ Wh,
                                               const float* __restrict__ scale,
                                               float* __restrict__ Whn, int n) {
    int i = blockIdx.x * 256 + threadIdx.x;
    if (i < n) Whn[i] = Wh[i] * scale[0];
}

// ---------------------------------------------------------------------------
// Kernel 3: gate projections. One wave per (t,b) row:
//   a_scale[t,b,k] = x_row . W_a[:,k] ;  rw[t,b,:] = softmax(x_row @ W_w)
// ---------------------------------------------------------------------------
__global__ __launch_bounds__(256) void k_gates(const float* __restrict__ x,
                                               const float* __restrict__ Wa,
                                               const float* __restrict__ Ww,
                                               float* __restrict__ ascale,
                                               float* __restrict__ rw) {
    const int wave = threadIdx.x >> 5;
    const int lane = threadIdx.x & 31;
    const int tb   = blockIdx.x * 8 + wave;      // < T*B
    const float* xr = x + (size_t)tb * DIM;

    float accA[KMEM] = {0.f, 0.f, 0.f, 0.f};
    float accW[KMEM] = {0.f, 0.f, 0.f, 0.f};
    for (int d = lane; d < DIM; d += 32) {
        float xv = xr[d];
#pragma unroll
        for (int k = 0; k < KMEM; ++k) {
            accA[k] += xv * Wa[d * KMEM + k];
            accW[k] += xv * Ww[d * KMEM + k];
        }
    }
#pragma unroll
    for (int k = 0; k < KMEM; ++k) {
        for (int off = 16; off > 0; off >>= 1) {
            accA[k] += __shfl_down(accA[k], off, 32);
            accW[k] += __shfl_down(accW[k], off, 32);
        }
    }
    if (lane == 0) {
        float mx = fmaxf(fmaxf(accW[0], accW[1]), fmaxf(accW[2], accW[3]));
        float e[KMEM], s = 0.0f;
#pragma unroll
        for (int k = 0; k < KMEM; ++k) { e[k] = expf(accW[k] - mx); s += e[k]; }
#pragma unroll
        for (int k = 0; k < KMEM; ++k) {
            ascale[(size_t)tb * KMEM + k] = accA[k];
            rw[(size_t)tb * KMEM + k]     = e[k] / s;
        }
    }
}

// ---------------------------------------------------------------------------
// Kernel 4: xproj = x @ W_x.T + b  as (T*B, D) = (T*B, D) x (D, D)^T
// via V_WMMA_F32_16X16X4_F32. One 16x16 tile per wave.
// A frag (16x4): lane l -> m = l&15, k = (l>>4)*2 + r  -> float2 load
// B frag (4x16): lane l -> n = l&15, k = (l>>4)*2 + r  -> W_x[n][k] float2
// C/D: lane l, vgpr r -> n = l&15, m = r + 8*(l>>4)
// ---------------------------------------------------------------------------
__global__ __launch_bounds__(256) void k_xproj(const float* __restrict__ X,
                                               const float* __restrict__ Wx,
                                               const float* __restrict__ bias,
                                               float* __restrict__ xproj) {
    const int wave = threadIdx.x >> 5;
    const int lane = threadIdx.x & 31;
    const int tile = blockIdx.x * 8 + wave;      // < (T*B/16)*(D/16)
    const int mt   = tile >> 6;                  // row tile (T*B/16 = 2048)
    const int nt   = tile & 63;                  // col tile (D/16 = 64)
    const int ml   = lane & 15;
    const int hi   = lane >> 4;

    const float* arow = X  + (size_t)(mt * 16 + ml) * DIM + 2 * hi;
    const float* brow = Wx + (size_t)(nt * 16 + ml) * DIM + 2 * hi;

    v8f acc = {0.f, 0.f, 0.f, 0.f, 0.f, 0.f, 0.f, 0.f};
#pragma unroll 8
    for (int k0 = 0; k0 < DIM; k0 += 4) {
        v2f a = *(const v2f*)(arow + k0);
        v2f b = *(const v2f*)(brow + k0);
        acc = __builtin_amdgcn_wmma_f32_16x16x4_f32(false, a, false, b,
                                                    (short)0, acc, false, false);
    }
    const float bv = bias[nt * 16 + ml];
    float* orow = xproj + (size_t)(mt * 16 + 8 * hi) * DIM + nt * 16 + ml;
#pragma unroll
    for (int r = 0; r < 8; ++r) orow[(size_t)r * DIM] = acc[r] + bv;
}

// ---------------------------------------------------------------------------
// Grid-wide generation barrier (counter pre-zeroed by hipMemsetAsync).
// ---------------------------------------------------------------------------
__device__ __forceinline__ void grid_sync(unsigned* counter, unsigned nblocks) {
    __threadfence();
    __syncthreads();
    if (threadIdx.x == 0) {
        unsigned my = atomicAdd(counter, 1u) + 1u;
        unsigned target = ((my + nblocks - 1u) / nblocks) * nblocks;
        while (atomicAdd(counter, 0u) < target) { __builtin_amdgcn_s_sleep(2); }
        __threadfence();
    }
    __syncthreads();
}

// ---------------------------------------------------------------------------
// Kernel 5: persistent sequential scan.
// 64 blocks x 256 threads (8 waves). Block owns cols [16*blk, 16*blk+16).
//   - W_hn column block (16 rows of W_hn = cols of W_hn.T) staged once in LDS
//     via the Tensor Data Mover, padded (+2 dwords / 256) for aligned,
//     low-conflict ds_load_b64 B-fragment reads.
//   - per step: 8 waves split K=1024 (128 each, 32 wmma f32 16x16x4),
//     partials reduced in LDS, + xproj, tanh -> h tile; m kept in LDS.
//   - one grid barrier per step; h double-buffered in global (L2-resident).
// ---------------------------------------------------------------------------
__global__ __launch_bounds__(256) void k_scan(const float* __restrict__ Whn,
                                              const float* __restrict__ xproj,
                                              const float* __restrict__ z,
                                              const float* __restrict__ h0,
                                              const float* __restrict__ m0,
                                              const float* __restrict__ a_g,
                                              const float* __restrict__ ascale,
                                              const float* __restrict__ rw,
                                              float* __restrict__ hbuf,   // [2][B][D]
                                              unsigned* __restrict__ counter,
                                              float* __restrict__ out) {
    extern __shared__ float smem[];
    float* Wlds = smem;                    // 16 * WROW
    float* plds = smem + 16 * WROW;        // 8 * 256 partial C tiles
    float* mlds = plds + 8 * 256;          // KMEM * 256 persistent m slice

    const int tidx = threadIdx.x;
    const int wave = tidx >> 5;
    const int lane = tidx & 31;
    const int ml   = lane & 15;
    const int hi   = lane >> 4;
    const int col0 = blockIdx.x * 16;
    const int b    = tidx >> 4;            // elementwise mapping: (b, n)
    const int n    = tidx & 15;
    const int d    = col0 + n;

    // ---- one-time TDM stage of the W block: 16 rows x 1024 f32, row-major,
    // pad 2 dwords every 256 -> LDS row stride WROW dwords ----
    if (wave == 0) {
        uint64_t ga = (uint64_t)(uintptr_t)(Whn + (size_t)col0 * DIM);
        unsigned lbase = __builtin_amdgcn_groupstaticsize();  // dynamic LDS base
        v4u g0;
        g0[0] = 1u;                                   // count = 1 valid D#
        g0[1] = lbase;                                // lds_addr (bytes)
        g0[2] = (unsigned)(ga & 0xFFFFFFFFull);       // global_addr[31:0]
        g0[3] = (unsigned)((ga >> 32) & 0x1FFFFFFull) // global_addr[56:32]
                | 0x80000000u;                        // type = 2 ("image")
        v8i g1;
        g1[0] = (int)((2u << 16)      // data_size = 4B
                    | (1u << 20)      // pad_enable
                    | (7u << 22)      // pad_interval: 256 dwords
                    | (1u << 25));    // pad_amount: 2 dwords
        g1[1] = (int)(1024u << 16);   // tensor_dim0 = 1024
        g1[2] = (int)(16u << 16);     // tensor_dim1 = 16
        g1[3] = (int)(1024u << 16);   // tile_dim0 = 1024
        g1[4] = (int)16u;             // tile_dim1 = 16
        g1[5] = (int)DIM;             // tensor_dim0_stride = 1024 elements
        g1[6] = 0;
        g1[7] = 0;
        v4i gz4 = {0, 0, 0, 0};                       // groups 2/3 disabled (2D)
        v8i gz8 = {0, 0, 0, 0, 0, 0, 0, 0};
        __builtin_amdgcn_tensor_load_to_lds(g0, g1, gz4, gz4, gz8, 0);
        __builtin_amdgcn_s_wait_tensorcnt(0);
    }

    // ---- init h[0] (this block's columns) and persistent m slice ----
    hbuf[(size_t)b * DIM + d] = h0[(size_t)b * DIM + d];
#pragma unroll
    for (int k = 0; k < KMEM; ++k)
        mlds[k * 256 + tidx] = m0[(size_t)k * BATCH * DIM + (size_t)b * DIM + d];

    grid_sync(counter, NBLK);   // h init + W stage visible everywhere

    int cur = 0;
    float hv = 0.0f;
    const int kbase = wave * 128;

    for (int t = 0; t < T_STEPS; ++t) {
        // ---- recurrent GEMM: D[b][n] += sum_k h[b][k] * W_hn[n][k] ----
        const float* hrow = hbuf + (size_t)cur * BATCH * DIM + (size_t)ml * DIM + 2 * hi;
        v8f acc = {0.f, 0.f, 0.f, 0.f, 0.f, 0.f, 0.f, 0.f};
#pragma unroll 8
        for (int kk = 0; kk < 32; ++kk) {
            int k = kbase + kk * 4 + 2 * hi;
            v2f a  = *(const v2f*)(hrow + kbase + kk * 4);
            v2f bb = *(const v2f*)(Wlds + ml * WROW + k + 2 * (k >> 8));
            acc = __builtin_amdgcn_wmma_f32_16x16x4_f32(false, a, false, bb,
                                                        (short)0, acc, false, false);
        }
#pragma unroll
        for (int r = 0; r < 8; ++r)
            plds[wave * 256 + (r + 8 * hi) * 16 + ml] = acc[r];
        __syncthreads();

        // ---- reduce partials, add xproj, tanh ----
        float s = 0.0f;
#pragma unroll
        for (int w = 0; w < 8; ++w) s += plds[w * 256 + tidx];
        const size_t row = (size_t)t * BATCH + b;
        hv = tanhf(xproj[row * DIM + d] + s);
        hbuf[(size_t)(cur ^ 1) * BATCH * DIM + (size_t)b * DIM + d] = hv;

        // ---- memory update + output (all state local to this block) ----
        float zh = z[row * 2 * DIM + d];
        float zm = z[row * 2 * DIM + DIM + d];
        float mem = 0.0f;
#pragma unroll
        for (int k = 0; k < KMEM; ++k) {
            float al = sigmoidf_(a_g[(size_t)k * DIM + d] + ascale[row * KMEM + k]);
            float mv = al * mlds[k * 256 + tidx] + (1.0f - al) * hv;
            mlds[k * 256 + tidx] = mv;
            mem += rw[row * KMEM + k] * mv;
        }
        out[row * DIM + d] = hv * siluf_(zh) + mem * siluf_(zm);

        grid_sync(counter, NBLK);
        cur ^= 1;
    }

    // ---- finals: h_f then m_f, appended after output in d_out ----
    float* hf = out + (size_t)T_STEPS * BATCH * DIM;
    hf[(size_t)b * DIM + d] = hv;
    float* mf = hf + (size_t)BATCH * DIM;
#pragma unroll
    for (int k = 0; k < KMEM; ++k)
        mf[(size_t)k * BATCH * DIM + (size_t)b * DIM + d] = mlds[k * 256 + tidx];
}

// ---------------------------------------------------------------------------
// Host launcher
// ---------------------------------------------------------------------------
extern "C" void kernel_launch(void* const* d_in, const int* in_sizes, int n_in,
                              void* d_out, int out_size, void* d_ws, size_t ws_size,
                              hipStream_t stream) {
    (void)in_sizes; (void)n_in; (void)out_size; (void)ws_size;
    const float* x    = (const float*)d_in[0];
    const float* z    = (const float*)d_in[1];
    const float* h0   = (const float*)d_in[2];
    const float* m0   = (const float*)d_in[3];
    const float* Wx   = (const float*)d_in[4];
    const float* Wh   = (const float*)d_in[5];
    const float* bias = (const float*)d_in[6];
    const float* a    = (const float*)d_in[7];
    const float* Wa   = (const float*)d_in[8];
    const float* Ww   = (const float*)d_in[9];
    float* out = (float*)d_out;

    // Workspace layout
    char* ws = (char*)d_ws;
    unsigned* counter = (unsigned*)ws;                           // 256 B
    float* scale  = (float*)(ws + 256);                          // 256 B
    float* Whn    = (float*)(ws + 512);                          // 4 MB
    float* hbuf   = (float*)(ws + 512 + (size_t)DIM * DIM * 4);  // 2*B*D
    float* ascale = hbuf + 2 * BATCH * DIM;                      // T*B*K
    float* rw     = ascale + (size_t)T_STEPS * BATCH * KMEM;     // T*B*K
    float* xproj  = rw + (size_t)T_STEPS * BATCH * KMEM;         // T*B*D (128 MB)

    (void)hipMemsetAsync(counter, 0, 256, stream);

    k_spectral<<<1, 1024, 0, stream>>>(Wh, scale);
    k_scale<<<(DIM * DIM + 255) / 256, 256, 0, stream>>>(Wh, scale, Whn, DIM * DIM);
    k_gates<<<(T_STEPS * BATCH) / 8, 256, 0, stream>>>(x, Wa, Ww, ascale, rw);
    k_xproj<<<((T_STEPS * BATCH / 16) * (DIM / 16)) / 8, 256, 0, stream>>>(x, Wx, bias, xproj);

    (void)hipFuncSetAttribute(reinterpret_cast<const void*>(k_scan),
                              hipFuncAttributeMaxDynamicSharedMemorySize, SCAN_SMEM_BYTES);
    k_scan<<<NBLK, 256, SCAN_SMEM_BYTES, stream>>>(Whn, xproj, z, h0, m0, a,
                                                   ascale, rw, hbuf, counter, out);
}